// PyramidRoIAlign2_68796786147715
// MI455X (gfx1250) — compile-verified
//
#include <hip/hip_runtime.h>

// ---------------------------------------------------------------------------
// Fused PyramidRoIAlign (4 levels, 7x7, sampling_ratio=1) + 1x1 conv reduce
// (K=1024 -> O=256) + bias + ReLU for MI455X (gfx1250, wave32).
//
// One 512-thread block (16 waves) per box. K is processed in 8 sub-chunks of
// 128 (level l = sc>>1, half = sc&1):
//   - weights sub-chunk staged to LDS *transposed* (pair-major B[k/2][o][2])
//     via GLOBAL_LOAD_ASYNC_TO_LDS_B64 (per-lane gather->scatter, ASYNCcnt),
//     8 lanes cover one full 64B line of w_reduce -> fully coalesced reads
//   - activations sub-chunk (bilinear gather, L2-resident pyramid) written
//     to LDS as A[k_local(128)][m(64)]
//   - GEMM: V_WMMA_F32_16X16X4_F32, each wave owns a 16(M)x64(O) strip,
//     A-fragment reused across 4 back-to-back WMMAs; B-frag = one
//     conflict-free ds_load_b64 (16 lanes -> 128 contiguous bytes)
// Accumulators (fp32, exact) persist across all 8 sub-chunks, then
// bias+ReLU, stage 256(o)x49(m) result in LDS, coalesced transposed store.
// ---------------------------------------------------------------------------

typedef __attribute__((ext_vector_type(2))) float v2f;
typedef __attribute__((ext_vector_type(8))) float v8f;

#define TPB   512
#define KTOT  1024
#define KC    128          // k sub-chunk
#define OOUT  256
#define MPAD  64
#define MREAL 49
#define BROW  518          // dwords per k-pair row of B in LDS (512 + 6 pad)

// LDS float layout:
//   [0 .. 8191]        A chunk [kc(128)][m(64)]            (32 KB)
//   [8192 .. 41343]    B chunk [pair(64)][BROW]            (~129.5 KB)
//   [0 .. 16639]       result  [o(256)][m pad 65]          (reuses A+B)
//   [41344 .. 41391]   coords: slx[8] sly[8] sx0[8] sx1[8] sy0[8] sy1[8]
#define A_OFF    0
#define B_OFF    8192
#define CRD_OFF  41344
#define SMEM_FLOATS 41392

__global__ __launch_bounds__(TPB)
void PyramidRoIAlign2_fused_wmma(const float* __restrict__ p0,
                                 const float* __restrict__ p1,
                                 const float* __restrict__ p2,
                                 const float* __restrict__ p3,
                                 const float* __restrict__ bboxes,
                                 const int*   __restrict__ binds,
                                 const float* __restrict__ wred,   // [256][1024]
                                 const float* __restrict__ bred,   // [256]
                                 float*       __restrict__ out)    // [N][256][7][7]
{
    extern __shared__ float smem[];
    float* Alds = smem + A_OFF;
    float* Blds = smem + B_OFF;
    float* slx  = smem + CRD_OFF;
    float* sly  = smem + CRD_OFF + 8;
    int*   sx0  = (int*)(smem + CRD_OFF + 16);
    int*   sx1  = (int*)(smem + CRD_OFF + 24);
    int*   sy0  = (int*)(smem + CRD_OFF + 32);
    int*   sy1  = (int*)(smem + CRD_OFF + 40);

    const int n    = blockIdx.x;
    const int tid  = threadIdx.x;
    const int lane = tid & 31;
    const int ww   = tid >> 5;          // wave id 0..15
    const int lr   = lane & 15;         // lane row 0..15
    const int hf   = lane >> 4;         // half-wave 0/1
    const int m0   = (ww & 3) * 16;     // M tile base (0,16,32,48)
    const int og   = ww >> 2;           // O group 0..3 -> o base og*64

    const int   bi  = binds[n];
    const float bx1 = bboxes[n * 4 + 0];
    const float by1 = bboxes[n * 4 + 1];
    const float bx2 = bboxes[n * 4 + 2];
    const float by2 = bboxes[n * 4 + 3];

    const float* pl[4] = { p0, p1, p2, p3 };
    const int    Hs[4] = { 192, 96, 48, 24 };
    const int    Ws[4] = { 256, 128, 64, 32 };
    const float  sc[4] = { 0.25f, 0.125f, 0.0625f, 0.03125f };

    v8f acc0 = {0.f,0.f,0.f,0.f,0.f,0.f,0.f,0.f};
    v8f acc1 = acc0, acc2 = acc0, acc3 = acc0;

#pragma unroll
    for (int s = 0; s < 8; ++s) {       // 8 K sub-chunks of 128
        const int l     = s >> 1;       // level
        const int half  = s & 1;        // channel half within level
        const int kbase = l * 256 + half * KC;
        const int Hl    = Hs[l];
        const int Wl    = Ws[l];

        __syncthreads();   // previous sub-chunk GEMM done reading Alds/Blds

        // --- per-level bin-center coordinates (7 x, 7 y) ---
        if (tid < 7) {
            const float scale = sc[l];
            const float x1s = bx1 * scale, x2s = bx2 * scale;
            const float y1s = by1 * scale, y2s = by2 * scale;
            const float bw = (x2s - x1s) * (1.0f / 7.0f);
            const float bh = (y2s - y1s) * (1.0f / 7.0f);
            float xs = x1s + ((float)tid + 0.5f) * bw;
            float ys = y1s + ((float)tid + 0.5f) * bh;
            xs = fminf(fmaxf(xs, 0.0f), (float)(Wl - 1));
            ys = fminf(fmaxf(ys, 0.0f), (float)(Hl - 1));
            const float x0f = floorf(xs);
            const float y0f = floorf(ys);
            slx[tid] = xs - x0f;
            sly[tid] = ys - y0f;
            const int x0i = (int)x0f;
            const int y0i = (int)y0f;
            sx0[tid] = x0i;
            sy0[tid] = y0i;
            sx1[tid] = min(x0i + 1, Wl - 1);
            sy1[tid] = min(y0i + 1, Hl - 1);
        }

        // --- stage B sub-chunk transposed into LDS via async gather->LDS ---
        // slot decomposition: 8 consecutive lanes read 8 consecutive k-pairs
        // (64B line) of one w_reduce row; LDS dest = Blds[p*BROW + o*2]
        {
            const float* wsrc = wred + kbase;
#pragma unroll 4
            for (int it = 0; it < (64 * OOUT) / TPB; ++it) {
                const int sl   = it * TPB + tid;       // 0..16383
                const int p8   = sl & 7;
                const int o    = (sl >> 3) & 255;
                const int pblk = sl >> 11;             // 0..7
                const int p    = pblk * 8 + p8;        // k-pair 0..63
                const float*  g    = wsrc + (size_t)o * KTOT + 2 * p;
                const unsigned ldsb =
                    (unsigned)(uintptr_t)(&Blds[p * BROW + o * 2]);
                asm volatile("global_load_async_to_lds_b64 %0, %1, off"
                             :: "v"(ldsb), "v"(g) : "memory");
            }
        }

        // --- gather: fill A[kc][m] (kc = channel within sub-chunk) ---
        {
            const float* fb = pl[l] +
                ((size_t)bi * OOUT + kbase - (size_t)l * 256 + half * 0) * 0 +
                (size_t)bi * 256 * Hl * Wl + (size_t)(half * KC) * Hl * Wl;
#pragma unroll 4
            for (int it = 0; it < (KC * MPAD) / TPB; ++it) {
                const int idx = it * TPB + tid;
                const int kc  = idx >> 6;        // channel 0..127
                const int m   = idx & 63;        // padded row 0..63
                float v = 0.0f;
                if (m < MREAL) {
                    const int h = m / 7;
                    const int w = m - 7 * h;
                    const float* fc = fb + (size_t)kc * (Hl * Wl);
                    const float lx = slx[w], ly = sly[h];
                    const int x0 = sx0[w], x1 = sx1[w];
                    const int y0 = sy0[h], y1 = sy1[h];
                    const float f00 = fc[y0 * Wl + x0];
                    const float f01 = fc[y0 * Wl + x1];
                    const float f10 = fc[y1 * Wl + x0];
                    const float f11 = fc[y1 * Wl + x1];
                    const float omx = 1.0f - lx, omy = 1.0f - ly;
                    v = f00 * omy * omx + f01 * omy * lx +
                        f10 * ly  * omx + f11 * ly  * lx;
                }
                Alds[kc * MPAD + m] = v;
            }
        }

        asm volatile("s_wait_asynccnt 0" ::: "memory");
        __syncthreads();

        // --- WMMA GEMM over this 128-wide K sub-chunk ---
        for (int kl = 0; kl < KC; kl += 4) {
            // A fragment: lanes 0-15 -> K = kl,kl+1 ; lanes 16-31 -> kl+2,kl+3
            v2f a;
            a.x = Alds[(kl + 2 * hf)     * MPAD + m0 + lr];
            a.y = Alds[(kl + 2 * hf + 1) * MPAD + m0 + lr];
            // B fragments: pair p = kl/2 + hf holds K = kl+2hf, kl+2hf+1
            const float* bp = &Blds[(kl / 2 + hf) * BROW + (og * 64 + lr) * 2];
            const v2f b0 = *(const v2f*)(bp);
            const v2f b1 = *(const v2f*)(bp + 32);
            const v2f b2 = *(const v2f*)(bp + 64);
            const v2f b3 = *(const v2f*)(bp + 96);
            acc0 = __builtin_amdgcn_wmma_f32_16x16x4_f32(false, a, false, b0,
                                                         (short)0, acc0, false, false);
            acc1 = __builtin_amdgcn_wmma_f32_16x16x4_f32(false, a, false, b1,
                                                         (short)0, acc1, false, false);
            acc2 = __builtin_amdgcn_wmma_f32_16x16x4_f32(false, a, false, b2,
                                                         (short)0, acc2, false, false);
            acc3 = __builtin_amdgcn_wmma_f32_16x16x4_f32(false, a, false, b3,
                                                         (short)0, acc3, false, false);
        }
    }

    __syncthreads();   // all waves done with Alds/Blds; reuse as result

    // --- bias + ReLU, stage result[o][m] in LDS (pad 65 -> conflict-free) ---
    float* Rlds = smem;
#pragma unroll
    for (int j = 0; j < 4; ++j) {
        const v8f cj = (j == 0) ? acc0 : (j == 1) ? acc1 : (j == 2) ? acc2 : acc3;
        const int o  = og * 64 + j * 16 + lr;
        const float bb = bred[o];
#pragma unroll
        for (int r = 0; r < 8; ++r) {
            const int m = m0 + r + 8 * hf;   // D layout: VGPR r, half hf -> M=r+8*hf
            float v = cj[r] + bb;
            v = (v > 0.0f) ? v : 0.0f;
            Rlds[o * 65 + m] = v;
        }
    }
    __syncthreads();

    // --- coalesced transposed store: out[n][o][m], m = h*7+w ---
    float* outn = out + (size_t)n * (OOUT * MREAL);
    for (int idx = tid; idx < OOUT * MREAL; idx += TPB) {
        const int o = idx / MREAL;
        const int m = idx - o * MREAL;
        outn[idx] = Rlds[o * 65 + m];
    }
}

extern "C" void kernel_launch(void* const* d_in, const int* in_sizes, int n_in,
                              void* d_out, int out_size, void* d_ws, size_t ws_size,
                              hipStream_t stream) {
    (void)n_in; (void)d_ws; (void)ws_size; (void)out_size;
    const float* p0     = (const float*)d_in[0];
    const float* p1     = (const float*)d_in[1];
    const float* p2     = (const float*)d_in[2];
    const float* p3     = (const float*)d_in[3];
    const float* bboxes = (const float*)d_in[4];
    const int*   binds  = (const int*)  d_in[5];
    const float* wred   = (const float*)d_in[6];
    const float* bred   = (const float*)d_in[7];
    float*       out    = (float*)d_out;

    const int N = in_sizes[4] / 4;                 // 1000 boxes
    const size_t smem_bytes = SMEM_FLOATS * sizeof(float);   // ~161.7 KB

    (void)hipFuncSetAttribute((const void*)PyramidRoIAlign2_fused_wmma,
                              hipFuncAttributeMaxDynamicSharedMemorySize,
                              (int)smem_bytes);

    PyramidRoIAlign2_fused_wmma<<<N, TPB, smem_bytes, stream>>>(
        p0, p1, p2, p3, bboxes, binds, wred, bred, out);
}